// Quantizer_83356725281204
// MI455X (gfx1250) — compile-verified
//
#include <hip/hip_runtime.h>

typedef __attribute__((ext_vector_type(2))) float v2f;
typedef __attribute__((ext_vector_type(8))) float v8f;

#define SROW 528  // float2 stride per packed-K row: 512 cols + 16 pad so the two
                  // 16-lane halves land on disjoint LDS bank ranges (b64 conflict-free)

// z: [32][64][32][32] fp32, codebook: [1024][64] fp32
// out: [2097152] quantized z (NCHW) + out[2097152]=commitment, out[2097153]=codebook loss
__global__ void __launch_bounds__(512) vq_kernel(const float* __restrict__ z,
                                                 const float* __restrict__ cb,
                                                 float* __restrict__ out,
                                                 float* __restrict__ partial)
{
    __shared__ float2 sCB[32 * SROW];   // 512 codebook rows, transposed, K pair-packed
    __shared__ float  sEsq[1024];       // 0.5*|e_n|^2
    __shared__ int    sIdx[16 * 16];    // per-wave winning indices
    __shared__ float  sRed[16];

    const int tid  = threadIdx.x;
    const int lane = tid & 31;
    const int wave = tid >> 5;
    const int tok  = lane & 15;   // M (tokens) for A, N (codes) for B/C/D
    const int hf   = lane >> 4;   // half-wave selects K sub-pair

    // ---- 0.5*|e|^2 table from global (codebook is L2-resident) ----
    for (int r = 0; r < 2; ++r) {
        int n = tid + r * 512;
        const float4* row = (const float4*)(cb + n * 64);
        float s = 0.f;
        #pragma unroll
        for (int q = 0; q < 16; ++q) {
            float4 v = row[q];
            s += v.x*v.x + v.y*v.y + v.z*v.z + v.w*v.w;
        }
        sEsq[n] = 0.5f * s;
    }

    // ---- load this wave's 16-token fp32 A tile (16x64), WMMA f32 A layout:
    //      lane(M)=tok, VGPR pair per K-step, halves hold K+{0,1} / K+{2,3} ----
    const int t0  = blockIdx.x * 256 + wave * 16;  // 16 waves * 16 tokens
    const int bB  = t0 >> 10;                      // batch (1024 tokens per image)
    const int hw0 = t0 & 1023;
    const float* abase = z + bB * 65536 + hw0 + tok;

    v2f a[16];
    #pragma unroll
    for (int kk = 0; kk < 16; ++kk) {
        int c0 = 4*kk + 2*hf;
        a[kk].x = abase[c0 * 1024];         // lanes 0-15 contiguous 64B per load
        a[kk].y = abase[(c0 + 1) * 1024];
    }

    float best[8];
    int   bidx[8];
    #pragma unroll
    for (int j = 0; j < 8; ++j) { best[j] = -3.4e38f; bidx[j] = 0; }

    const float4* cb4 = (const float4*)cb;

    for (int phase = 0; phase < 2; ++phase) {
        __syncthreads();   // previous phase fully consumed
        // stage 512 rows transposed: sCB[k2][n] = (cb[n][2*k2], cb[n][2*k2+1])
        #pragma unroll
        for (int k4 = 0; k4 < 16; ++k4) {
            float4 v = cb4[(phase * 512 + tid) * 16 + k4];
            sCB[(2*k4    ) * SROW + tid] = make_float2(v.x, v.y);  // b64, stride-8B: no conflicts
            sCB[(2*k4 + 1) * SROW + tid] = make_float2(v.z, v.w);
        }
        __syncthreads();

        for (int c16 = 0; c16 < 32; ++c16) {
            int nl = c16 * 16 + tok;          // local column in staged half
            int ng = phase * 512 + nl;        // global codebook index for this lane
            float he = sEsq[ng];
            v8f acc = {};
            #pragma unroll
            for (int kk = 0; kk < 16; ++kk) { // K=64 via 16 chained fp32 WMMAs
                float2 bq = sCB[(2*kk + hf) * SROW + nl];
                v2f bv; bv.x = bq.x; bv.y = bq.y;
                acc = __builtin_amdgcn_wmma_f32_16x16x4_f32(
                    false, a[kk], false, bv, (short)0, acc, false, false);
            }
            // score = x.e - 0.5|e|^2 ; argmin(dist) == argmax(score)
            #pragma unroll
            for (int j = 0; j < 8; ++j) {
                float s = acc[j] - he;
                if (s > best[j]) { best[j] = s; bidx[j] = ng; }  // strict > keeps lowest n
            }
        }
    }

    // ---- reduce argmax across the 16 lanes of each half (tie -> lower index) ----
    #pragma unroll
    for (int j = 0; j < 8; ++j) {
        #pragma unroll
        for (int off = 8; off >= 1; off >>= 1) {
            float os = __shfl_xor(best[j], off, 16);
            int   oi = __shfl_xor(bidx[j], off, 16);
            if (os > best[j] || (os == best[j] && oi < bidx[j])) { best[j] = os; bidx[j] = oi; }
        }
    }
    if (lane == 0) {
        #pragma unroll
        for (int j = 0; j < 8; ++j) sIdx[wave*16 + j] = bidx[j];       // tokens 0..7
    }
    if (lane == 16) {
        #pragma unroll
        for (int j = 0; j < 8; ++j) sIdx[wave*16 + 8 + j] = bidx[j];   // tokens 8..15
    }
    int myIdx = sIdx[wave*16 + tok];  // same-wave LDS store->load, in order

    // ---- gather winning row, write quantized output, accumulate (q-z)^2 ----
    const float* crow = cb + myIdx * 64;
    float* obase = out + bB * 65536 + hw0 + tok;
    float part = 0.f;
    #pragma unroll
    for (int kk = 0; kk < 16; ++kk) {
        int c0 = 4*kk + 2*hf;
        float2 q = *(const float2*)(crow + c0);
        float dx = q.x - a[kk].x;
        float dy = q.y - a[kk].y;
        part += dx*dx + dy*dy;
        obase[c0 * 1024]       = q.x;   // 16 contiguous floats per half-wave per channel
        obase[(c0 + 1) * 1024] = q.y;
    }

    #pragma unroll
    for (int off = 16; off >= 1; off >>= 1) part += __shfl_xor(part, off, 32);
    if (lane == 0) sRed[wave] = part;
    __syncthreads();
    if (tid == 0) {
        float s = 0.f;
        #pragma unroll
        for (int w = 0; w < 16; ++w) s += sRed[w];
        partial[blockIdx.x] = s;
    }
}

__global__ void vq_finalize(const float* __restrict__ partial, float* __restrict__ out)
{
    __shared__ float s[4];
    int tid = threadIdx.x;
    float v = partial[tid];   // 128 block partials
    #pragma unroll
    for (int off = 16; off >= 1; off >>= 1) v += __shfl_xor(v, off, 32);
    if ((tid & 31) == 0) s[tid >> 5] = v;
    __syncthreads();
    if (tid == 0) {
        float t = s[0] + s[1] + s[2] + s[3];
        float m = t * (1.0f / 2097152.0f);   // mean over B*HW*C
        out[2097152] = m;   // commitment loss
        out[2097153] = m;   // codebook loss (numerically identical)
    }
}

extern "C" void kernel_launch(void* const* d_in, const int* in_sizes, int n_in,
                              void* d_out, int out_size, void* d_ws, size_t ws_size,
                              hipStream_t stream) {
    (void)in_sizes; (void)n_in; (void)out_size; (void)ws_size;
    const float* z  = (const float*)d_in[0];   // [32,64,32,32] fp32
    const float* cb = (const float*)d_in[1];   // [1024,64] fp32
    float* out      = (float*)d_out;
    float* partial  = (float*)d_ws;            // 128 floats of scratch

    vq_kernel<<<128, 512, 0, stream>>>(z, cb, out, partial);
    vq_finalize<<<1, 128, 0, stream>>>(partial, out);
}